// HATDecoder_8048768713445
// MI455X (gfx1250) — compile-verified
//
#include <hip/hip_runtime.h>
#include <hip/hip_bf16.h>
#include <stdint.h>

typedef __attribute__((ext_vector_type(2))) float v2f;
typedef __attribute__((ext_vector_type(8))) float v8f;

#define NROWS 8192
#define DIM   128
#define CLS   16
#define ALPHA 0.2f
#define SPLIT 16      // waves per 16-row tile (j-dimension split)
#define JCHUNK (NROWS / SPLIT)   // 512 columns per wave

__device__ __forceinline__ float leaky(float x) { return x > 0.0f ? x : ALPHA * x; }

// Monotone float -> uint key (for atomicMax over floats of any sign)
__device__ __forceinline__ unsigned fkey(float f) {
    unsigned u = __float_as_uint(f);
    return (u & 0x80000000u) ? ~u : (u | 0x80000000u);
}
__device__ __forceinline__ float fdecode(unsigned k) {
    unsigned u = (k & 0x80000000u) ? (k & 0x7FFFFFFFu) : ~k;
    return __uint_as_float(u);
}

// ---------------- init: zero L, O accumulators and the maxkey ----------------
__global__ void hat_init(unsigned* p, int n) {
    int i = blockIdx.x * blockDim.x + threadIdx.x;
    if (i < n) p[i] = 0u;
}

// ---------------- phase 1: h = logmap0(x)@W + b ; s_src, s_dst, max(s_dst) ----
__global__ void hat_phase1(const float* __restrict__ x, const float* __restrict__ W,
                           const float* __restrict__ b, const float* __restrict__ a_src,
                           const float* __restrict__ a_dst,
                           float* __restrict__ h, float* __restrict__ ssrc,
                           float* __restrict__ sdst, unsigned* __restrict__ maxkey) {
    __shared__ float sW[DIM * CLS];
    __shared__ float sB[CLS], sAs[CLS], sAd[CLS];
    int tid = threadIdx.x;
    for (int i = tid; i < DIM * CLS; i += blockDim.x) sW[i] = W[i];
    if (tid < CLS) { sB[tid] = b[tid]; sAs[tid] = a_src[tid]; sAd[tid] = a_dst[tid]; }
    __syncthreads();

    int row = blockIdx.x * blockDim.x + tid;
    if (row >= NROWS) return;

    const float* xr = x + (size_t)row * DIM;
    float acc[CLS];
#pragma unroll
    for (int c = 0; c < CLS; ++c) acc[c] = 0.0f;
    float nrm2 = 0.0f;
    for (int d = 0; d < DIM; ++d) {
        float xd = xr[d];
        nrm2 = fmaf(xd, xd, nrm2);
#pragma unroll
        for (int c = 0; c < CLS; ++c) acc[c] = fmaf(xd, sW[d * CLS + c], acc[c]);
    }
    float norm = fmaxf(sqrtf(nrm2), 1e-15f);
    float an = fminf(norm, 1.0f - 1e-7f);            // artanh clip (norm >= 0)
    float at = 0.5f * (log1pf(an) - log1pf(-an));    // artanh(norm)
    float scale = at / norm;

    float ss = 0.0f, sd = 0.0f;
#pragma unroll
    for (int c = 0; c < CLS; ++c) {
        float hv = fmaf(scale, acc[c], sB[c]);
        h[(size_t)row * CLS + c] = hv;
        ss = fmaf(hv, sAs[c], ss);
        sd = fmaf(hv, sAd[c], sd);
    }
    ssrc[row] = ss;
    sdst[row] = sd;
    atomicMax(maxkey, fkey(sd));
}

// ---------------- phase 2: masked-softmax attention + WMMA aggregation -------
__global__ void hat_phase2(const uint8_t* __restrict__ adj, const float* __restrict__ h,
                           const float* __restrict__ ssrc, const float* __restrict__ sdst,
                           const unsigned* __restrict__ maxkey,
                           float* __restrict__ L, float* __restrict__ O) {
    int wid   = (blockIdx.x * blockDim.x + threadIdx.x) >> 5;   // global wave id
    int lane  = threadIdx.x & 31;
    int tile  = wid >> 4;          // 0..511 : 16-row tile
    int split = wid & (SPLIT - 1); // 0..15  : j-chunk
    int m  = lane & 15;
    int hf = lane >> 4;            // lane half: 0 -> K 0/1, 1 -> K 2/3
    int row = tile * 16 + m;

    float sr  = ssrc[row];
    float mx  = fdecode(*maxkey);
    float mrow = leaky(sr + mx);   // safe shift >= true row max (softmax-invariant)

    const uint8_t* arow = adj + (size_t)row * NROWS;
    v8f acc = {};
    float lsum = 0.0f;

    int jbeg = split * JCHUNK;
    int jend = jbeg + JCHUNK;
#pragma unroll 4
    for (int jb = jbeg; jb < jend; jb += 4) {
        int j0 = jb + hf * 2;
        unsigned short mm = *(const unsigned short*)(arow + j0);     // 2 adj bytes
        float2 sd2 = *(const float2*)(sdst + j0);                    // 2 s_dst
        float e0 = leaky(sr + sd2.x) - mrow;
        float e1 = leaky(sr + sd2.y) - mrow;
        float p0 = (mm & 0x00FFu) ? __expf(e0) : 0.0f;
        float p1 = (mm & 0xFF00u) ? __expf(e1) : 0.0f;
        lsum += p0 + p1;
        v2f A = {p0, p1};
        v2f B = {h[(size_t)j0 * CLS + m], h[(size_t)(j0 + 1) * CLS + m]};
        acc = __builtin_amdgcn_wmma_f32_16x16x4_f32(
            /*neg_a=*/false, A, /*neg_b=*/false, B,
            /*c_mod=*/(short)0, acc, /*reuse_a=*/false, /*reuse_b=*/false);
    }

    // row-sum of p lives split across lane halves (lane m and m+16)
    lsum += __shfl_xor(lsum, 16);
    if (hf == 0) atomicAdd(&L[row], lsum);

    // C layout: lane l, vgpr v -> (M = v + 8*(l>=16), N = l&15)
#pragma unroll
    for (int v = 0; v < 8; ++v) {
        int M = v + 8 * hf;
        atomicAdd(&O[((size_t)tile * 16 + M) * CLS + m], acc[v]);
    }
}

// ---------------- phase 3: normalize, elu, expmap0, proj ---------------------
__global__ void hat_phase3(const float* __restrict__ O, const float* __restrict__ L,
                           float* __restrict__ out) {
    int row = blockIdx.x * blockDim.x + threadIdx.x;
    if (row >= NROWS) return;
    float inv = 1.0f / L[row];
    float u[CLS];
    float nrm2 = 0.0f;
#pragma unroll
    for (int c = 0; c < CLS; ++c) {
        float v = O[(size_t)row * CLS + c] * inv;
        float e = (v > 0.0f) ? v : expm1f(v);       // elu
        u[c] = e;
        nrm2 = fmaf(e, e, nrm2);
    }
    float norm = fmaxf(sqrtf(nrm2), 1e-15f);
    float sc = tanhf(norm) / norm;                  // expmap0 (c=1)
    float y[CLS];
    float ynrm2 = 0.0f;
#pragma unroll
    for (int c = 0; c < CLS; ++c) { y[c] = sc * u[c]; ynrm2 = fmaf(y[c], y[c], ynrm2); }
    float ynorm = fmaxf(sqrtf(ynrm2), 1e-15f);
    const float maxnorm = 1.0f - 1e-5f;
    float f = (ynorm > maxnorm) ? (maxnorm / ynorm) : 1.0f;
#pragma unroll
    for (int c = 0; c < CLS; ++c) out[(size_t)row * CLS + c] = y[c] * f;
}

extern "C" void kernel_launch(void* const* d_in, const int* in_sizes, int n_in,
                              void* d_out, int out_size, void* d_ws, size_t ws_size,
                              hipStream_t stream) {
    const float*   x     = (const float*)d_in[0];
    const uint8_t* adj   = (const uint8_t*)d_in[1];   // jnp bool -> 1 byte/elem
    const float*   W     = (const float*)d_in[2];
    const float*   b     = (const float*)d_in[3];
    const float*   a_src = (const float*)d_in[4];
    const float*   a_dst = (const float*)d_in[5];
    float* out = (float*)d_out;

    // workspace layout (floats)
    float* ws   = (float*)d_ws;
    float* h    = ws;                        // 8192*16
    float* ssrc = h + NROWS * CLS;           // 8192
    float* sdst = ssrc + NROWS;              // 8192
    float* L    = sdst + NROWS;              // 8192
    float* O    = L + NROWS;                 // 8192*16
    unsigned* maxkey = (unsigned*)(O + NROWS * CLS); // 1

    // zero L, O, maxkey (contiguous region starting at L)
    int zn = NROWS + NROWS * CLS + 1;
    hat_init<<<(zn + 255) / 256, 256, 0, stream>>>((unsigned*)L, zn);

    hat_phase1<<<NROWS / 256, 256, 0, stream>>>(x, W, b, a_src, a_dst,
                                                h, ssrc, sdst, maxkey);

    // 512 tiles * 16 splits = 8192 waves -> 1024 blocks of 256 (8 waves)
    int nwaves = (NROWS / 16) * SPLIT;
    hat_phase2<<<(nwaves * 32) / 256, 256, 0, stream>>>(adj, h, ssrc, sdst,
                                                        maxkey, L, O);

    hat_phase3<<<NROWS / 256, 256, 0, stream>>>(O, L, out);
}